// DotProductSelfAttentionLayer_77970836292246
// MI455X (gfx1250) — compile-verified
//
#include <hip/hip_runtime.h>
#include <math.h>

#define ATT_B 4
#define ATT_N 4096
#define ATT_D 128
#define KTILE 32
#define NTILES (ATT_N / KTILE)

typedef float v2f __attribute__((ext_vector_type(2)));
typedef float v8f __attribute__((ext_vector_type(8)));

#define WMMA_F32(A, B, C) \
    __builtin_amdgcn_wmma_f32_16x16x4_f32(false, (A), false, (B), (short)0, (C), false, false)

// gfx1250 async global->LDS copy (ASYNCcnt-tracked), GV addressing mode.
__device__ __forceinline__ void async_load_b128(unsigned lds_off, const float* gaddr) {
    asm volatile("global_load_async_to_lds_b128 %0, %1, off"
                 :: "v"(lds_off), "v"(gaddr) : "memory");
}
__device__ __forceinline__ void wait_async0() {
    asm volatile("s_wait_asynccnt 0x0" ::: "memory");
}

// Flash attention, fp32 WMMA (V_WMMA_F32_16X16X4_F32), wave32.
// One wave = 16 query rows; one block = 4 waves = 64 query rows.
// K/V tiles (32x128) stream into LDS via double-buffered async copies.
__global__ __launch_bounds__(128, 1)
void attn_fwd_f32wmma(const float* __restrict__ x, float* __restrict__ out)
{
    const int tid  = threadIdx.x;
    const int wave = tid >> 5;
    const int lane = tid & 31;
    const int half = lane >> 4;   // 0: lanes 0-15, 1: lanes 16-31
    const int l16  = lane & 15;

    const int b  = blockIdx.y;
    const int q0 = blockIdx.x * 64 + wave * 16;  // first query row of this wave

    __shared__ float kbuf[2][KTILE][ATT_D];  // 2 x 16 KB double-buffered K/V tiles
    __shared__ float pbuf[4][16][36];        // per-wave P transpose scratch (pad 36)

    const float* xb = x + (size_t)b * ATT_N * ATT_D;

    const unsigned lds0 = (unsigned)(uintptr_t)&kbuf[0][0][0];
    const unsigned lds1 = (unsigned)(uintptr_t)&kbuf[1][0][0];

    // kick off async copy of K-tile 0 into kbuf[0]: 4096 floats = 1024 b128 chunks
    #pragma unroll
    for (int j = 0; j < 8; ++j) {
        const int chunk = tid + 128 * j;
        async_load_b128(lds0 + chunk * 16, xb + chunk * 4);
    }

    // ---- Q A-fragments: 16x128 tile as 32 fragments of 16x4 f32 ----
    // lane holds Q[q0 + l16][4f + 2*half + {0,1}]  (float2, contiguous)
    v2f qa[32];
    {
        const float* qbase = xb + (size_t)(q0 + l16) * ATT_D + 2 * half;
        #pragma unroll
        for (int f = 0; f < 32; ++f)
            qa[f] = *(const v2f*)(qbase + 4 * f);
    }

    // O accumulator: 8 tiles of 16x16 f32 (cols 16t + l16, rows r + 8*half)
    v8f o[8];
    #pragma unroll
    for (int t = 0; t < 8; ++t) o[t] = {};

    float mrow[8], lrow[8];
    #pragma unroll
    for (int r = 0; r < 8; ++r) { mrow[r] = -INFINITY; lrow[r] = 0.0f; }

    for (int it = 0; it < NTILES; ++it) {
        wait_async0();    // this wave's issued tile transfers have landed in LDS
        __syncthreads();  // all waves' transfers visible; prev buffer free

        if (it + 1 < NTILES) {   // start streaming the next tile into the other buffer
            const float* g = xb + (size_t)(it + 1) * KTILE * ATT_D;
            const unsigned ldsn = (it & 1) ? lds0 : lds1;
            #pragma unroll
            for (int j = 0; j < 8; ++j) {
                const int chunk = tid + 128 * j;
                async_load_b128(ldsn + chunk * 16, g + chunk * 4);
            }
        }
        const float (*kt)[ATT_D] = kbuf[it & 1];

        // ---- S = Q * K^T : two 16x16 accumulators (keys 0-15 / 16-31 of tile) ----
        // Software-pipelined B-fragment loads: group of 4 K-steps in flight ahead.
        v8f s0 = {}, s1 = {};
        {
            v2f c0[4], c1[4];
            #pragma unroll
            for (int u = 0; u < 4; ++u) {
                c0[u] = *(const v2f*)(&kt[l16][4 * u + 2 * half]);
                c1[u] = *(const v2f*)(&kt[16 + l16][4 * u + 2 * half]);
            }
            #pragma unroll
            for (int g = 0; g < 8; ++g) {
                v2f n0[4], n1[4];
                if (g < 7) {
                    #pragma unroll
                    for (int u = 0; u < 4; ++u) {
                        const int f = 4 * (g + 1) + u;
                        n0[u] = *(const v2f*)(&kt[l16][4 * f + 2 * half]);
                        n1[u] = *(const v2f*)(&kt[16 + l16][4 * f + 2 * half]);
                    }
                }
                #pragma unroll
                for (int u = 0; u < 4; ++u) {
                    s0 = WMMA_F32(qa[4 * g + u], c0[u], s0);
                    s1 = WMMA_F32(qa[4 * g + u], c1[u], s1);
                }
                if (g < 7) {
                    #pragma unroll
                    for (int u = 0; u < 4; ++u) { c0[u] = n0[u]; c1[u] = n1[u]; }
                }
            }
        }

        // ---- online softmax over 32 keys (row r = query r + 8*half, lanes = key) ----
        float scale[8];
        #pragma unroll
        for (int r = 0; r < 8; ++r) {
            float v = fmaxf(s0[r], s1[r]);
            #pragma unroll
            for (int off = 1; off < 16; off <<= 1)
                v = fmaxf(v, __shfl_xor(v, off, 32));
            float mnew = fmaxf(mrow[r], v);
            scale[r] = __expf(mrow[r] - mnew);
            mrow[r]  = mnew;
            float p0 = __expf(s0[r] - mnew);
            float p1 = __expf(s1[r] - mnew);
            s0[r] = p0; s1[r] = p1;
            float sum = p0 + p1;
            #pragma unroll
            for (int off = 1; off < 16; off <<= 1)
                sum += __shfl_xor(sum, off, 32);
            lrow[r] = lrow[r] * scale[r] + sum;
        }

        // rescale O accumulators by per-row correction factor
        #pragma unroll
        for (int t = 0; t < 8; ++t) {
            #pragma unroll
            for (int r = 0; r < 8; ++r)
                o[t][r] *= scale[r];
        }

        // ---- move P (16x32) from C-layout to A-layout through per-wave LDS ----
        #pragma unroll
        for (int r = 0; r < 8; ++r) {
            pbuf[wave][r + 8 * half][l16]      = s0[r];   // pbuf[query][key]
            pbuf[wave][r + 8 * half][16 + l16] = s1[r];
        }
        __builtin_amdgcn_wave_barrier();  // DS is in-order per wave; just stop reordering

        v2f pa[8];
        #pragma unroll
        for (int f = 0; f < 8; ++f) {
            pa[f][0] = pbuf[wave][l16][4 * f + 2 * half + 0];
            pa[f][1] = pbuf[wave][l16][4 * f + 2 * half + 1];
        }

        // ---- O += P * V : 8 column tiles x 8 K-steps, pipelined V-fragment loads ----
        {
            v2f cv[8];
            #pragma unroll
            for (int f = 0; f < 8; ++f) {
                cv[f][0] = kt[4 * f + 2 * half + 0][l16];
                cv[f][1] = kt[4 * f + 2 * half + 1][l16];
            }
            #pragma unroll
            for (int t = 0; t < 8; ++t) {
                v2f nv[8];
                if (t < 7) {
                    #pragma unroll
                    for (int f = 0; f < 8; ++f) {
                        nv[f][0] = kt[4 * f + 2 * half + 0][16 * (t + 1) + l16];
                        nv[f][1] = kt[4 * f + 2 * half + 1][16 * (t + 1) + l16];
                    }
                }
                #pragma unroll
                for (int f = 0; f < 8; ++f)
                    o[t] = WMMA_F32(pa[f], cv[f], o[t]);
                if (t < 7) {
                    #pragma unroll
                    for (int f = 0; f < 8; ++f) cv[f] = nv[f];
                }
            }
        }
    }

    // ---- normalize by softmax denominator and store ----
    float* ob = out + (size_t)b * ATT_N * ATT_D;
    #pragma unroll
    for (int r = 0; r < 8; ++r) {
        const float inv = 1.0f / lrow[r];
        float* orow = ob + (size_t)(q0 + r + 8 * half) * ATT_D + l16;
        #pragma unroll
        for (int t = 0; t < 8; ++t)
            orow[16 * t] = o[t][r] * inv;
    }
}

extern "C" void kernel_launch(void* const* d_in, const int* in_sizes, int n_in,
                              void* d_out, int out_size, void* d_ws, size_t ws_size,
                              hipStream_t stream) {
    (void)in_sizes; (void)n_in; (void)out_size; (void)d_ws; (void)ws_size;
    const float* x = (const float*)d_in[0];
    float* out = (float*)d_out;
    dim3 grid(ATT_N / 64, ATT_B, 1);
    attn_fwd_f32wmma<<<grid, dim3(128, 1, 1), 0, stream>>>(x, out);
}